// NenwinModel_63548336111988
// MI455X (gfx1250) — compile-verified
//
#include <hip/hip_runtime.h>

typedef __attribute__((ext_vector_type(2))) float v2f;
typedef __attribute__((ext_vector_type(8))) float v8f;

#define N_NODES   1024
#define N_TOT     4096
#define N_MARBLES 3072
#define N_EATERS  64
#define EPS       1e-6f

// ---------------------------------------------------------------------------
// Kernel 1: all-pairs force + Euler integration.
// One wave32 owns a 16-row tile of i. For each 16-column chunk of j it
// computes the 16x16 Gram tile p_i . p_j with one V_WMMA_F32_16X16X4_F32,
// forms dist^2 via |p_i|^2 + |p_j|^2 - 2*Gram, and accumulates
//   Sx = sum_j w*x_j, Sy = sum_j w*y_j, S1 = sum_j w,   w = m_j*(d2+eps)^-1.5
// per lane (each lane holds a fixed column n = lane&15 of the C tile, rows
// rbase..rbase+7). acc_i = (Sx - x_i*S1, Sy - y_i*S1)  (m_i cancels).
// The i==j mask is only applied in the single diagonal chunk (uniform branch).
// rsqrt uses the raw v_rsq_f32 (argument >= 1e-6, no denorm guard needed).
// ---------------------------------------------------------------------------
__global__ __launch_bounds__(128) void nbody_step_kernel(
    const float* __restrict__ pos,
    const float* __restrict__ vel,
    const float* __restrict__ mass,
    const float* __restrict__ dtp,
    float* __restrict__ out)            // [2][N_TOT][2] flattened
{
  const int lane = threadIdx.x & 31;
  const int wave = threadIdx.x >> 5;
  const int tile = blockIdx.x * 4 + wave;      // 0..255
  const int i0   = tile * 16;

  const int  n  = lane & 15;                   // column within tile
  const bool hi = lane >= 16;

  // A operand (16x4, K in VGPRs): lanes 0-15 carry K=0,1 = (x_i, y_i);
  // lanes 16-31 carry K=2,3 = 0 padding.
  float axl = pos[2 * (i0 + n) + 0];
  float ayl = pos[2 * (i0 + n) + 1];
  v2f A;
  A.x = hi ? 0.0f : axl;
  A.y = hi ? 0.0f : ayl;

  // Rows covered by this lane's 8 C elements.
  const int rbase = hi ? 8 : 0;
  float ri[8];
#pragma unroll
  for (int k = 0; k < 8; ++k) {
    int r   = i0 + rbase + k;
    float x = pos[2 * r + 0];
    float y = pos[2 * r + 1];
    ri[k]   = x * x + y * y;
  }

  float Sx[8], Sy[8], S1[8];
#pragma unroll
  for (int k = 0; k < 8; ++k) { Sx[k] = 0.f; Sy[k] = 0.f; S1[k] = 0.f; }

#pragma unroll 2
  for (int c = 0; c < N_TOT / 16; ++c) {
    const int j = c * 16 + n;                  // this lane's column body
    float pjx = pos[2 * j + 0];
    float pjy = pos[2 * j + 1];
    float mj  = mass[j];
    float rj  = pjx * pjx + pjy * pjy;

    // prefetch next chunk (speculative; emits global_prefetch_b8)
    __builtin_prefetch(&pos[2 * (c * 16 + 16 + n)], 0, 1);
    __builtin_prefetch(&mass[c * 16 + 16 + n], 0, 1);

    // B operand (4x16, K in VGPRs): lanes 0-15 K=0,1 = (x_j, y_j); hi -> 0.
    v2f B;
    B.x = hi ? 0.0f : pjx;
    B.y = hi ? 0.0f : pjy;

    v8f C = {};
    C = __builtin_amdgcn_wmma_f32_16x16x4_f32(
        /*neg_a=*/false, A, /*neg_b=*/false, B,
        /*c_mod=*/(short)0, C, /*reuse_a=*/false, /*reuse_b=*/false);

    float w[8];
#pragma unroll
    for (int k = 0; k < 8; ++k) {
      float g  = C[k];
      float d2 = ri[k] + rj - 2.0f * g;        // dist^2 via Gram identity
      float t  = d2 + EPS;                     // >= 1e-6: raw v_rsq is safe
      float rs = __builtin_amdgcn_rsqf(t);
      float wv = mj * rs * rs * rs;
      w[k] = (d2 > 0.0f) ? wv : 0.0f;
    }

    if (c == tile) {                           // uniform branch: diagonal chunk
#pragma unroll
      for (int k = 0; k < 8; ++k)
        if ((rbase + k) == n) w[k] = 0.0f;     // mask the i==j self-pair
    }

#pragma unroll
    for (int k = 0; k < 8; ++k) {
      Sx[k] += w[k] * pjx;
      Sy[k] += w[k] * pjy;
      S1[k] += w[k];
    }
  }

  // 16-lane butterfly reduction (xor masks <16 keep the two halves separate).
#pragma unroll
  for (int k = 0; k < 8; ++k) {
#pragma unroll
    for (int m = 1; m <= 8; m <<= 1) {
      Sx[k] += __shfl_xor(Sx[k], m, 32);
      Sy[k] += __shfl_xor(Sy[k], m, 32);
      S1[k] += __shfl_xor(S1[k], m, 32);
    }
  }

  // lanes 0 and 16 now hold full sums for rows rbase..rbase+7.
  if (n == 0) {
    float dt = dtp[0];
#pragma unroll
    for (int k = 0; k < 8; ++k) {
      int r   = i0 + rbase + k;
      float x = pos[2 * r + 0];
      float y = pos[2 * r + 1];
      float ax = Sx[k] - x * S1[k];            // m_i cancels in acc
      float ay = Sy[k] - y * S1[k];
      float vx = vel[2 * r + 0] + ax * dt;
      float vy = vel[2 * r + 1] + ay * dt;
      float nx = x + vx * dt;
      float ny = y + vy * dt;
      out[2 * r + 0] = nx;
      out[2 * r + 1] = ny;
      out[2 * N_TOT + 2 * r + 0] = vx;
      out[2 * N_TOT + 2 * r + 1] = vy;
    }
  }
}

// ---------------------------------------------------------------------------
// Kernel 2: eaten test. Reads pos_new from d_out (stream-ordered after k1).
// ---------------------------------------------------------------------------
__global__ __launch_bounds__(256) void eaten_kernel(
    const float* __restrict__ state,           // pos_new at state[0..2*N_TOT)
    const int*   __restrict__ eater_idx,
    const float* __restrict__ eater_radius,
    float* __restrict__ eaten)                 // N_MARBLES floats
{
  __shared__ float ex[N_EATERS], ey[N_EATERS], er2[N_EATERS];
  const int t = threadIdx.x;
  if (t < N_EATERS) {
    int ei = eater_idx[t];
    ex[t]  = state[2 * ei + 0];
    ey[t]  = state[2 * ei + 1];
    float r = eater_radius[t];
    er2[t] = r * r;
  }
  __syncthreads();

  const int m = blockIdx.x * blockDim.x + t;
  if (m < N_MARBLES) {
    const int g = N_NODES + m;
    float x = state[2 * g + 0];
    float y = state[2 * g + 1];
    bool any = false;
#pragma unroll 8
    for (int e = 0; e < N_EATERS; ++e) {
      float dx = x - ex[e];
      float dy = y - ey[e];
      any = any || (dx * dx + dy * dy <= er2[e]);
    }
    eaten[m] = any ? 1.0f : 0.0f;
  }
}

extern "C" void kernel_launch(void* const* d_in, const int* in_sizes, int n_in,
                              void* d_out, int out_size, void* d_ws, size_t ws_size,
                              hipStream_t stream) {
  const float* positions    = (const float*)d_in[0];  // (4096,2)
  const float* velocities   = (const float*)d_in[1];  // (4096,2)
  const float* masses       = (const float*)d_in[2];  // (4096,)
  const int*   eater_idx    = (const int*)  d_in[3];  // (64,)
  const float* eater_radius = (const float*)d_in[4];  // (64,)
  const float* dt           = (const float*)d_in[5];  // (1,)

  float* out   = (float*)d_out;                // [0,16384) = state_new
  float* eaten = out + 2 * 2 * N_TOT;          // [16384, 19456) = eaten

  // 256 waves (one per 16-row tile), 4 waves per block.
  nbody_step_kernel<<<N_TOT / 16 / 4, 128, 0, stream>>>(
      positions, velocities, masses, dt, out);

  eaten_kernel<<<(N_MARBLES + 255) / 256, 256, 0, stream>>>(
      out, eater_idx, eater_radius, eaten);
}